// PAM_Module_86406152061377
// MI455X (gfx1250) — compile-verified
//
#include <hip/hip_runtime.h>

// PAM attention map: out[b,n,m] = softmax_m( sum_c (Wq x + bq)[b,c,n] * (Wk x + bk)[b,c,m] )
// B=4, C=5, H=W=96, N=9216.  Output 4*9216*9216 f32 = 1.36 GB  -> pure store-bandwidth bound.
// Energy GEMM uses V_WMMA_F32_16X16X4_F32 (full fp32, K=4), channels padded 5 -> 8 (two chained WMMAs).

#define B_  4
#define C_  5
#define N_  9216          // 96*96
#define CP  8             // padded channel count (2 x K=4 WMMA)
#define NT16   (N_/16)    // 576 row/col tiles
#define NT128  (N_/128)   // 72  column blocks of 8 waves

typedef __attribute__((ext_vector_type(2))) float v2f;
typedef __attribute__((ext_vector_type(8))) float v8f;

__device__ __forceinline__ v8f wmma_f32_k4(v2f a, v2f b, v8f c) {
    // D = A(16x4 f32) * B(4x16 f32) + C(16x16 f32)
    return __builtin_amdgcn_wmma_f32_16x16x4_f32(
        /*neg_a=*/false, a, /*neg_b=*/false, b,
        /*c_mod=*/(short)0, c, /*reuse_a=*/false, /*reuse_b=*/false);
}

// ---------------------------------------------------------------------------
// Pass 1: 1x1 convs -> Q,K stored channel-LAST, padded to 8 channels (zeros).
// Qp/Kp layout: [b][n][8]  (32-byte aligned rows -> WMMA operands are b64 loads)
// ---------------------------------------------------------------------------
__global__ __launch_bounds__(256) void pam_proj(
    const float* __restrict__ x,
    const float* __restrict__ wq, const float* __restrict__ bq,
    const float* __restrict__ wk, const float* __restrict__ bk,
    float* __restrict__ Qp, float* __restrict__ Kp)
{
    int idx = blockIdx.x * 256 + threadIdx.x;          // b*N + n
    if (idx >= B_ * N_) return;
    int b = idx / N_;
    int n = idx - b * N_;

    float xv[C_];
#pragma unroll
    for (int c = 0; c < C_; ++c)
        xv[c] = x[(size_t)(b * C_ + c) * N_ + n];

    float q[C_], k[C_];
#pragma unroll
    for (int o = 0; o < C_; ++o) {
        float aq = bq[o], ak = bk[o];
#pragma unroll
        for (int c = 0; c < C_; ++c) {
            aq = fmaf(wq[o * C_ + c], xv[c], aq);
            ak = fmaf(wk[o * C_ + c], xv[c], ak);
        }
        q[o] = aq; k[o] = ak;
    }

    float4* qo = (float4*)(Qp + (size_t)idx * CP);
    float4* ko = (float4*)(Kp + (size_t)idx * CP);
    qo[0] = make_float4(q[0], q[1], q[2], q[3]);
    qo[1] = make_float4(q[4], 0.f, 0.f, 0.f);          // zero pad channels 5..7
    ko[0] = make_float4(k[0], k[1], k[2], k[3]);
    ko[1] = make_float4(k[4], 0.f, 0.f, 0.f);
}

// Compute one 16x16 energy tile: rows n0..n0+15 (A fixed per block), cols m0..m0+15.
__device__ __forceinline__ v8f energy_tile(const float* __restrict__ Kp,
                                           int b, int m0, int l15, int kb,
                                           v2f aLo, v2f aHi)
{
    const float* kcol = Kp + ((size_t)b * N_ + m0 + l15) * CP;
    v2f bLo = { kcol[kb],     kcol[kb + 1] };          // K = kb, kb+1   (ch 0..3)
    v2f bHi = { kcol[4 + kb], kcol[5 + kb] };          // K' = kb, kb+1  (ch 4..7)
    v8f acc = {};
    acc = wmma_f32_k4(aLo, bLo, acc);
    acc = wmma_f32_k4(aHi, bHi, acc);
    return acc;
}

// ---------------------------------------------------------------------------
// Pass 2: per-row max, then per-row sum(exp(e - max)).  One block = (b, 16 rows).
// 8 waves sweep the 9216 columns in 16-wide tiles (72 iterations each).
// ---------------------------------------------------------------------------
__global__ __launch_bounds__(256) void pam_stats(
    const float* __restrict__ Qp, const float* __restrict__ Kp,
    float* __restrict__ rMax, float* __restrict__ rSum)
{
    const int b    = blockIdx.x / NT16;
    const int n0   = (blockIdx.x % NT16) * 16;
    const int lane = threadIdx.x & 31;
    const int wave = threadIdx.x >> 5;
    const int half = lane >> 4;                        // 0: K/M 0..,  1: upper half
    const int l15  = lane & 15;
    const int kb   = half * 2;

    // A operand (fixed across the column sweep): A[M=l15][K] = Q[ch][n0+l15]
    const float* qrow = Qp + ((size_t)b * N_ + n0 + l15) * CP;
    v2f aLo = { qrow[kb],     qrow[kb + 1] };
    v2f aHi = { qrow[4 + kb], qrow[5 + kb] };

    __shared__ float red[8 * 16];
    __shared__ float rowm[16];

    // ---- phase A: row max (no exp) ----
    float mx[8];
#pragma unroll
    for (int r = 0; r < 8; ++r) mx[r] = -3.0e38f;

    for (int it = 0; it < NT128; ++it) {
        const int m0 = (it * 8 + wave) * 16;
        v8f acc = energy_tile(Kp, b, m0, l15, kb, aLo, aHi);
#pragma unroll
        for (int r = 0; r < 8; ++r) mx[r] = fmaxf(mx[r], acc[r]);
    }
    // butterfly within 16-lane halves: lanes 0-15 carry rows 0..7, 16-31 rows 8..15
#pragma unroll
    for (int d = 1; d < 16; d <<= 1)
#pragma unroll
        for (int r = 0; r < 8; ++r) mx[r] = fmaxf(mx[r], __shfl_xor(mx[r], d, 32));
    if (l15 == 0) {
#pragma unroll
        for (int r = 0; r < 8; ++r) red[wave * 16 + half * 8 + r] = mx[r];
    }
    __syncthreads();
    if (threadIdx.x < 16) {
        float m = -3.0e38f;
        for (int w = 0; w < 8; ++w) m = fmaxf(m, red[w * 16 + threadIdx.x]);
        rowm[threadIdx.x] = m;
    }
    __syncthreads();

    float rm[8];
#pragma unroll
    for (int r = 0; r < 8; ++r) rm[r] = rowm[half * 8 + r];

    // ---- phase B: sum of exp(e - rowmax), 1 exp per element ----
    float sm[8];
#pragma unroll
    for (int r = 0; r < 8; ++r) sm[r] = 0.f;

    for (int it = 0; it < NT128; ++it) {
        const int m0 = (it * 8 + wave) * 16;
        v8f acc = energy_tile(Kp, b, m0, l15, kb, aLo, aHi);
#pragma unroll
        for (int r = 0; r < 8; ++r) sm[r] += __expf(acc[r] - rm[r]);
    }
#pragma unroll
    for (int d = 1; d < 16; d <<= 1)
#pragma unroll
        for (int r = 0; r < 8; ++r) sm[r] += __shfl_xor(sm[r], d, 32);
    if (l15 == 0) {
#pragma unroll
        for (int r = 0; r < 8; ++r) red[wave * 16 + half * 8 + r] = sm[r];
    }
    __syncthreads();
    if (threadIdx.x < 16) {
        float s = 0.f;
        for (int w = 0; w < 8; ++w) s += red[w * 16 + threadIdx.x];
        rMax[(size_t)b * N_ + n0 + threadIdx.x] = rowm[threadIdx.x];
        rSum[(size_t)b * N_ + n0 + threadIdx.x] = s;
    }
}

// ---------------------------------------------------------------------------
// Pass 3: recompute energy tiles and stream out exp(e - max) / sum.
// One block = (b, 16 rows, 128 cols); each wave owns one 16x16 tile.
// This is the only pass that touches the 1.36 GB output.
// ---------------------------------------------------------------------------
__global__ __launch_bounds__(256) void pam_write(
    const float* __restrict__ Qp, const float* __restrict__ Kp,
    const float* __restrict__ rMax, const float* __restrict__ rSum,
    float* __restrict__ out)
{
    int t = blockIdx.x;
    const int mb = t % NT128;  t /= NT128;
    const int nb = t % NT16;
    const int b  = t / NT16;
    const int n0 = nb * 16;

    const int lane = threadIdx.x & 31;
    const int wave = threadIdx.x >> 5;
    const int half = lane >> 4;
    const int l15  = lane & 15;
    const int kb   = half * 2;
    const int m0   = mb * 128 + wave * 16;

    __shared__ float smax[16], sinv[16];
    if (threadIdx.x < 16) {
        smax[threadIdx.x] = rMax[(size_t)b * N_ + n0 + threadIdx.x];
        sinv[threadIdx.x] = 1.0f / rSum[(size_t)b * N_ + n0 + threadIdx.x];
    }

    const float* qrow = Qp + ((size_t)b * N_ + n0 + l15) * CP;
    v2f aLo = { qrow[kb],     qrow[kb + 1] };
    v2f aHi = { qrow[4 + kb], qrow[5 + kb] };

    v8f acc = energy_tile(Kp, b, m0, l15, kb, aLo, aHi);

    __syncthreads();

    const int rbase = half * 8;                        // C/D layout: lanes 16-31 hold M=8..15
    const size_t obase = ((size_t)b * N_ + n0) * (size_t)N_ + (size_t)(m0 + l15);
#pragma unroll
    for (int r = 0; r < 8; ++r) {
        const int row = rbase + r;
        out[obase + (size_t)row * N_] = __expf(acc[r] - smax[row]) * sinv[row];
    }
}

// ---------------------------------------------------------------------------
extern "C" void kernel_launch(void* const* d_in, const int* in_sizes, int n_in,
                              void* d_out, int out_size, void* d_ws, size_t ws_size,
                              hipStream_t stream) {
    const float* x  = (const float*)d_in[0];
    const float* wq = (const float*)d_in[1];
    const float* bq = (const float*)d_in[2];
    const float* wk = (const float*)d_in[3];
    const float* bk = (const float*)d_in[4];
    float* out = (float*)d_out;

    // Workspace layout (floats): Qp[B*N*8] | Kp[B*N*8] | rMax[B*N] | rSum[B*N]  (~2.65 MB)
    float* Qp   = (float*)d_ws;
    float* Kp   = Qp + (size_t)B_ * N_ * CP;
    float* rMax = Kp + (size_t)B_ * N_ * CP;
    float* rSum = rMax + (size_t)B_ * N_;

    pam_proj <<<(B_ * N_ + 255) / 256, 256, 0, stream>>>(x, wq, bq, wk, bk, Qp, Kp);
    pam_stats<<<B_ * NT16,            256, 0, stream>>>(Qp, Kp, rMax, rSum);
    pam_write<<<B_ * NT16 * NT128,    256, 0, stream>>>(Qp, Kp, rMax, rSum, out);
}